// EnhancedFusionGate_43885975830852
// MI455X (gfx1250) — compile-verified
//
#include <hip/hip_runtime.h>
#include <hip/hip_bf16.h>
#include <math.h>

#define NPTS 16384
#define NC   96
#define KTOP 16
#define KNM  8

typedef __attribute__((ext_vector_type(2))) float v2f;
typedef __attribute__((ext_vector_type(8))) float v8f;

// ---------------------------------------------------------------------------
// KNN: one thread per point, xyz tiled through LDS, register top-16 insertion.
// d2 = |p|^2 + |q|^2 - 2 p.q  (exactly the reference formula)
// ---------------------------------------------------------------------------
__global__ __launch_bounds__(256)
void knn_kernel(const float* __restrict__ xyz, int* __restrict__ knn_idx)
{
    __shared__ float sx[2048], sy[2048], sz[2048];
    const int i = blockIdx.x * 256 + threadIdx.x;
    const float px = xyz[i * 3 + 0];
    const float py = xyz[i * 3 + 1];
    const float pz = xyz[i * 3 + 2];
    const float ps = px * px + py * py + pz * pz;

    float bd[KTOP];
    int   bi[KTOP];
#pragma unroll
    for (int s = 0; s < KTOP; ++s) { bd[s] = 3.4e38f; bi[s] = 0; }

    for (int t = 0; t < NPTS; t += 2048) {
        for (int q = threadIdx.x; q < 2048; q += 256) {
            sx[q] = xyz[(t + q) * 3 + 0];
            sy[q] = xyz[(t + q) * 3 + 1];
            sz[q] = xyz[(t + q) * 3 + 2];
        }
        __syncthreads();
        for (int q = 0; q < 2048; ++q) {
            const float qx = sx[q], qy = sy[q], qz = sz[q];
            const float qs = qx * qx + qy * qy + qz * qz;
            const float d2 = ps + qs - 2.0f * (px * qx + py * qy + pz * qz);
            if (d2 < bd[KTOP - 1]) {
                float cd = d2; int ci = t + q;
#pragma unroll
                for (int s = 0; s < KTOP; ++s) {
                    if (cd < bd[s]) {
                        float tf = bd[s]; bd[s] = cd; cd = tf;
                        int   ti = bi[s]; bi[s] = ci; ci = ti;
                    }
                }
            }
        }
        __syncthreads();
    }
#pragma unroll
    for (int s = 0; s < KTOP; ++s) knn_idx[i * KTOP + s] = bi[s];
}

// ---------------------------------------------------------------------------
// Build base_input (N,192), fusion_input (N,192), cons_input cols [0,192).
// One thread per (point, channel).
// ---------------------------------------------------------------------------
__global__ __launch_bounds__(256)
void build_inputs_kernel(const float* __restrict__ f2d, const float* __restrict__ f3d,
                         const int* __restrict__ knn_idx,
                         float* __restrict__ base_in, float* __restrict__ fus_in,
                         float* __restrict__ cons_in)
{
    const int tid = blockIdx.x * 256 + threadIdx.x;
    const int i = tid / NC;
    const int c = tid - i * NC;
    const float f2 = f2d[i * NC + c];
    const float f3 = f3d[i * NC + c];
    float s2 = 0.0f, s3 = 0.0f;
#pragma unroll
    for (int k = 0; k < KTOP; ++k) {
        const int j = knn_idx[i * KTOP + k];
        s2 += f2d[j * NC + c];
        s3 += f3d[j * NC + c];
    }
    base_in[i * 192 + c]      = f2;
    base_in[i * 192 + 96 + c] = f3;
    cons_in[i * 208 + c]      = f2;
    cons_in[i * 208 + 96 + c] = f3;
    fus_in[i * 192 + c]       = f2 + s2 * (1.0f / 16.0f);
    fus_in[i * 192 + 96 + c]  = f3 + s3 * (1.0f / 16.0f);
}

// ---------------------------------------------------------------------------
// Normals (3x3 eigh, smallest eigenvector) + geo_encoder MLP (6->32->16 relu),
// writes cons_input cols [192,208). One thread per point.
// ---------------------------------------------------------------------------
__global__ __launch_bounds__(256)
void normals_geo_kernel(const float* __restrict__ xyz, const int* __restrict__ knn_idx,
                        const float* __restrict__ w0, const float* __restrict__ b0, // (32,6),(32)
                        const float* __restrict__ w1, const float* __restrict__ b1, // (16,32),(16)
                        float* __restrict__ cons_in)
{
    const int i = blockIdx.x * 256 + threadIdx.x;
    const float cx = xyz[i * 3 + 0], cy = xyz[i * 3 + 1], cz = xyz[i * 3 + 2];

    float a00 = 0, a01 = 0, a02 = 0, a11 = 0, a12 = 0, a22 = 0;
#pragma unroll
    for (int k = 0; k < KNM; ++k) {
        const int j = knn_idx[i * KTOP + 1 + k];
        const float dx = xyz[j * 3 + 0] - cx;
        const float dy = xyz[j * 3 + 1] - cy;
        const float dz = xyz[j * 3 + 2] - cz;
        a00 += dx * dx; a01 += dx * dy; a02 += dx * dz;
        a11 += dy * dy; a12 += dy * dz; a22 += dz * dz;
    }

    // smallest eigenvalue via trigonometric method
    const float q  = (a00 + a11 + a22) * (1.0f / 3.0f);
    const float p1 = a01 * a01 + a02 * a02 + a12 * a12;
    const float d0 = a00 - q, d1 = a11 - q, d2 = a22 - q;
    const float p2 = d0 * d0 + d1 * d1 + d2 * d2 + 2.0f * p1;
    const float p  = sqrtf(p2 * (1.0f / 6.0f));
    float lmin = q;
    if (p > 1e-20f) {
        const float inv = 1.0f / p;
        const float c00 = d0 * inv, c11 = d1 * inv, c22 = d2 * inv;
        const float c01 = a01 * inv, c02 = a02 * inv, c12 = a12 * inv;
        float r = 0.5f * (c00 * (c11 * c22 - c12 * c12)
                        - c01 * (c01 * c22 - c12 * c02)
                        + c02 * (c01 * c12 - c11 * c02));
        r = fminf(1.0f, fmaxf(-1.0f, r));
        const float phi = acosf(r) * (1.0f / 3.0f);
        lmin = q + 2.0f * p * cosf(phi + 2.0943951023931953f); // smallest
    }

    // eigenvector of (A - lmin I) via row cross products
    const float r0x = a00 - lmin, r0y = a01,        r0z = a02;
    const float r1x = a01,        r1y = a11 - lmin, r1z = a12;
    const float r2x = a02,        r2y = a12,        r2z = a22 - lmin;
    float vx = r0y * r1z - r0z * r1y;
    float vy = r0z * r1x - r0x * r1z;
    float vz = r0x * r1y - r0y * r1x;
    float nn = vx * vx + vy * vy + vz * vz;
    if (nn < 1e-24f) {
        vx = r0y * r2z - r0z * r2y;
        vy = r0z * r2x - r0x * r2z;
        vz = r0x * r2y - r0y * r2x;
        nn = vx * vx + vy * vy + vz * vz;
    }
    if (nn < 1e-24f) {
        vx = r1y * r2z - r1z * r2y;
        vy = r1z * r2x - r1x * r2z;
        vz = r1x * r2y - r1y * r2x;
        nn = vx * vx + vy * vy + vz * vz;
    }
    if (nn < 1e-24f) { vx = 1.0f; vy = 0.0f; vz = 0.0f; nn = 1.0f; }
    const float rcp = 1.0f / fmaxf(sqrtf(nn), 1e-12f);
    vx *= rcp; vy *= rcp; vz *= rcp;

    const float g[6] = { cx, cy, cz, vx, vy, vz };
    float h[32];
#pragma unroll
    for (int o = 0; o < 32; ++o) {
        float acc = b0[o];
#pragma unroll
        for (int k = 0; k < 6; ++k) acc += w0[o * 6 + k] * g[k];
        h[o] = fmaxf(acc, 0.0f);
    }
#pragma unroll
    for (int o = 0; o < 16; ++o) {
        float acc = b1[o];
#pragma unroll
        for (int k = 0; k < 32; ++k) acc += w1[k + o * 32] * h[k];
        cons_in[i * 208 + 192 + o] = fmaxf(acc, 0.0f);
    }
}

// ---------------------------------------------------------------------------
// Dense layer on the f32 WMMA pipe: Y = act(X @ W^T + b).
// One wave computes a 32x16 output block as two 16x16 WMMA tiles that share
// the B operand (per K-step: 2 A loads + 1 B load -> 2x V_WMMA_F32_16X16X4_F32).
// Full column tiles take the unmasked fast loop (no per-WMMA VALU); only the
// edge tile (dout % 16) pays the B mask. EXEC stays all-1s for every WMMA
// (branches are wave-uniform), per the ISA restriction.
// ACT: 0=relu, 1=sigmoid, 2=identity
// ---------------------------------------------------------------------------
template <int ACT>
__global__ __launch_bounds__(256)
void mlp_wmma_kernel(const float* __restrict__ X, const float* __restrict__ W,
                     const float* __restrict__ Bv, float* __restrict__ Y,
                     int din, int dout, int ldx, int ldy)
{
    const int lane = threadIdx.x & 31;
    const int wave = blockIdx.x * (blockDim.x >> 5) + (threadIdx.x >> 5);
    const int ct   = (dout + 15) >> 4;
    const int row_pair = wave / ct;             // 32 rows per wave
    const int col_tile = wave - row_pair * ct;
    if (row_pair >= (NPTS >> 5)) return;        // uniform per wave

    const int lo = lane & 15;
    const int hi = lane >> 4;
    const int m0 = (row_pair << 5) + lo;        // A row, tile 0
    const int m1 = m0 + 16;                     // A row, tile 1
    const int n  = (col_tile << 4) + lo;        // B col this lane feeds

    const float* __restrict__ xrow0 = X + (size_t)m0 * ldx;
    const float* __restrict__ xrow1 = X + (size_t)m1 * ldx;

    v8f acc0 = {};
    v8f acc1 = {};

    if ((col_tile << 4) + 16 <= dout) {
        // full tile: unmasked B
        const float* __restrict__ wrow = W + (size_t)n * din;
        for (int k0 = 0; k0 < din; k0 += 4) {
            const int ka = k0 + (hi << 1);
            v2f a0, a1, b;
            a0.x = xrow0[ka]; a0.y = xrow0[ka + 1];
            a1.x = xrow1[ka]; a1.y = xrow1[ka + 1];
            b.x  = wrow[ka];  b.y  = wrow[ka + 1];
            acc0 = __builtin_amdgcn_wmma_f32_16x16x4_f32(false, a0, false, b,
                                                         (short)0, acc0, false, false);
            acc1 = __builtin_amdgcn_wmma_f32_16x16x4_f32(false, a1, false, b,
                                                         (short)0, acc1, false, false);
        }
    } else {
        // edge tile: mask B columns >= dout (values zeroed, EXEC untouched)
        const float bmask = (n < dout) ? 1.0f : 0.0f;
        const int   ncl   = (n < dout) ? n : 0;
        const float* __restrict__ wrow = W + (size_t)ncl * din;
        for (int k0 = 0; k0 < din; k0 += 4) {
            const int ka = k0 + (hi << 1);
            v2f a0, a1, b;
            a0.x = xrow0[ka]; a0.y = xrow0[ka + 1];
            a1.x = xrow1[ka]; a1.y = xrow1[ka + 1];
            b.x  = wrow[ka] * bmask; b.y = wrow[ka + 1] * bmask;
            acc0 = __builtin_amdgcn_wmma_f32_16x16x4_f32(false, a0, false, b,
                                                         (short)0, acc0, false, false);
            acc1 = __builtin_amdgcn_wmma_f32_16x16x4_f32(false, a1, false, b,
                                                         (short)0, acc1, false, false);
        }
    }

    if (n < dout) {
        const float bias = Bv[n];
#pragma unroll
        for (int r = 0; r < 8; ++r) {
            const int orow = (row_pair << 5) + r + (hi << 3);
            float v0 = acc0[r] + bias;
            float v1 = acc1[r] + bias;
            if (ACT == 0)      { v0 = fmaxf(v0, 0.0f);             v1 = fmaxf(v1, 0.0f); }
            else if (ACT == 1) { v0 = 1.0f / (1.0f + expf(-v0));   v1 = 1.0f / (1.0f + expf(-v1)); }
            Y[(size_t)orow * ldy + n]        = v0;
            Y[(size_t)(orow + 16) * ldy + n] = v1;
        }
    }
}

// ---------------------------------------------------------------------------
// gate_fusion MLP (3->16->1 sigmoid) + mask + feature blend + confidence.
// One thread per point.
// ---------------------------------------------------------------------------
__global__ __launch_bounds__(256)
void finalize_kernel(const float* __restrict__ f2d, const float* __restrict__ f3d,
                     const int* __restrict__ valid,
                     const float* __restrict__ g_base, const float* __restrict__ g_sp,
                     const float* __restrict__ g_geo, const float* __restrict__ conf_raw,
                     const float* __restrict__ w0, const float* __restrict__ b0, // (16,3),(16)
                     const float* __restrict__ w1, const float* __restrict__ b1, // (1,16),(1)
                     float* __restrict__ out)
{
    const int i = blockIdx.x * 256 + threadIdx.x;
    const float g[3] = { g_base[i], g_sp[i], g_geo[i] };
    float h[16];
#pragma unroll
    for (int o = 0; o < 16; ++o) {
        float acc = b0[o];
#pragma unroll
        for (int k = 0; k < 3; ++k) acc += w0[o * 3 + k] * g[k];
        h[o] = fmaxf(acc, 0.0f);
    }
    float acc = b1[0];
#pragma unroll
    for (int k = 0; k < 16; ++k) acc += w1[k] * h[k];
    float fg = 1.0f / (1.0f + expf(-acc));

    const float vw = (valid[i] != 0) ? 1.0f : 0.0f;
    fg = fg * vw + fg * 0.1f * (1.0f - vw);

    for (int c = 0; c < NC; ++c) {
        const float f2 = f2d[i * NC + c];
        const float f3 = f3d[i * NC + c];
        out[i * NC + c] = fg * f2 + (1.0f - fg) * f3;
    }
    out[(size_t)NPTS * NC + i] = conf_raw[i] * (vw * 0.9f + 0.1f);
}

// ---------------------------------------------------------------------------
// Host launcher
// ---------------------------------------------------------------------------
extern "C" void kernel_launch(void* const* d_in, const int* in_sizes, int n_in,
                              void* d_out, int out_size, void* d_ws, size_t ws_size,
                              hipStream_t stream)
{
    (void)in_sizes; (void)n_in; (void)out_size; (void)ws_size;
    const float* f2d   = (const float*)d_in[0];
    const float* f3d   = (const float*)d_in[1];
    const float* xyz   = (const float*)d_in[2];
    const int*   valid = (const int*)d_in[3];
    // params flattened in setup_inputs() insertion order, (w,b) per layer
    const float* bg_w0 = (const float*)d_in[4];  const float* bg_b0 = (const float*)d_in[5];
    const float* bg_w1 = (const float*)d_in[6];  const float* bg_b1 = (const float*)d_in[7];
    const float* bg_w2 = (const float*)d_in[8];  const float* bg_b2 = (const float*)d_in[9];
    const float* sa_w0 = (const float*)d_in[10]; const float* sa_b0 = (const float*)d_in[11];
    const float* sa_w1 = (const float*)d_in[12]; const float* sa_b1 = (const float*)d_in[13];
    const float* sa_w2 = (const float*)d_in[14]; const float* sa_b2 = (const float*)d_in[15];
    const float* ge_w0 = (const float*)d_in[16]; const float* ge_b0 = (const float*)d_in[17];
    const float* ge_w1 = (const float*)d_in[18]; const float* ge_b1 = (const float*)d_in[19];
    const float* co_w0 = (const float*)d_in[20]; const float* co_b0 = (const float*)d_in[21];
    const float* co_w1 = (const float*)d_in[22]; const float* co_b1 = (const float*)d_in[23];
    const float* co_w2 = (const float*)d_in[24]; const float* co_b2 = (const float*)d_in[25];
    const float* gf_w0 = (const float*)d_in[26]; const float* gf_b0 = (const float*)d_in[27];
    const float* gf_w1 = (const float*)d_in[28]; const float* gf_b1 = (const float*)d_in[29];
    const float* cf_w0 = (const float*)d_in[30]; const float* cf_b0 = (const float*)d_in[31];
    const float* cf_w1 = (const float*)d_in[32]; const float* cf_b1 = (const float*)d_in[33];

    float* out = (float*)d_out;

    // workspace carve-up
    float* base_in = (float*)d_ws;                  // N*192
    float* fus_in  = base_in + (size_t)NPTS * 192;  // N*192
    float* cons_in = fus_in  + (size_t)NPTS * 192;  // N*208
    float* h1      = cons_in + (size_t)NPTS * 208;  // N*64
    float* h2      = h1      + (size_t)NPTS * 64;   // N*64
    float* g_base  = h2      + (size_t)NPTS * 64;   // N
    float* g_sp    = g_base  + NPTS;                // N
    float* g_geo   = g_sp    + NPTS;                // N
    float* c_raw   = g_geo   + NPTS;                // N
    int*   knn_idx = (int*)(c_raw + NPTS);          // N*16 ints

    const int ptBlocks = NPTS / 256;

    knn_kernel<<<ptBlocks, 256, 0, stream>>>(xyz, knn_idx);
    build_inputs_kernel<<<(NPTS * NC) / 256, 256, 0, stream>>>(
        f2d, f3d, knn_idx, base_in, fus_in, cons_in);
    normals_geo_kernel<<<ptBlocks, 256, 0, stream>>>(
        xyz, knn_idx, ge_w0, ge_b0, ge_w1, ge_b1, cons_in);

    auto mlp_blocks = [](int dout) {
        const int ct = (dout + 15) / 16;
        const int waves = (NPTS / 32) * ct;   // 32 rows per wave
        return (waves * 32 + 255) / 256;
    };

    // base_gate: 192->64 relu, 64->32 relu, 32->1 sigmoid
    mlp_wmma_kernel<0><<<mlp_blocks(64), 256, 0, stream>>>(base_in, bg_w0, bg_b0, h1, 192, 64, 192, 64);
    mlp_wmma_kernel<0><<<mlp_blocks(32), 256, 0, stream>>>(h1, bg_w1, bg_b1, h2, 64, 32, 64, 32);
    mlp_wmma_kernel<1><<<mlp_blocks(1),  256, 0, stream>>>(h2, bg_w2, bg_b2, g_base, 32, 1, 32, 1);

    // spatial_attn: 192->64 relu, 64->64 relu, 64->1 sigmoid
    mlp_wmma_kernel<0><<<mlp_blocks(64), 256, 0, stream>>>(fus_in, sa_w0, sa_b0, h1, 192, 64, 192, 64);
    mlp_wmma_kernel<0><<<mlp_blocks(64), 256, 0, stream>>>(h1, sa_w1, sa_b1, h2, 64, 64, 64, 64);
    mlp_wmma_kernel<1><<<mlp_blocks(1),  256, 0, stream>>>(h2, sa_w2, sa_b2, g_sp, 64, 1, 64, 1);

    // consistency: 208->64 relu, 64->32 relu, 32->1 sigmoid
    mlp_wmma_kernel<0><<<mlp_blocks(64), 256, 0, stream>>>(cons_in, co_w0, co_b0, h1, 208, 64, 208, 64);
    mlp_wmma_kernel<0><<<mlp_blocks(32), 256, 0, stream>>>(h1, co_w1, co_b1, h2, 64, 32, 64, 32);
    mlp_wmma_kernel<1><<<mlp_blocks(1),  256, 0, stream>>>(h2, co_w2, co_b2, g_geo, 32, 1, 32, 1);

    // confidence: 192->32 relu, 32->1 sigmoid
    mlp_wmma_kernel<0><<<mlp_blocks(32), 256, 0, stream>>>(base_in, cf_w0, cf_b0, h1, 192, 32, 192, 32);
    mlp_wmma_kernel<1><<<mlp_blocks(1),  256, 0, stream>>>(h1, cf_w1, cf_b1, c_raw, 32, 1, 32, 1);

    finalize_kernel<<<ptBlocks, 256, 0, stream>>>(
        f2d, f3d, valid, g_base, g_sp, g_geo, c_raw,
        gf_w0, gf_b0, gf_w1, gf_b1, out);
}